// IF_49340584296832
// MI455X (gfx1250) — compile-verified
//
#include <hip/hip_runtime.h>

// Integrate-and-fire scan: y[t] = (v+x[t] >= 1), v = (v+x[t]) - y[t]
// Shapes: x,y are (T=32, B=32, N=65536) f32. Columns (B*N) are independent;
// only T is sequential. Memory-bound: ~537 MB traffic -> ~23us @ 23.3 TB/s.
// Data path: 8-deep global_load_async_to_lds_b128 pipeline -> ds_load_b128
// -> VALU IF step -> nontemporal global_store_b128.

typedef float v4f __attribute__((ext_vector_type(4)));
// Builtin parameter types (confirmed by hipcc diagnostic in round 1):
//   param0: '__attribute__((__vector_size__(4*sizeof(int)))) int *' (generic)
//   param1: same element type, address_space(3)
typedef int v4i_gcc __attribute__((__vector_size__(4 * sizeof(int))));
typedef __attribute__((address_space(3))) v4i_gcc lds_v4i;

#define T_STEPS 32
#define BLOCK   256
#define VEC     4
#define DEPTH   8   // async pipeline depth (power of 2)

static constexpr unsigned long long BN = 32ull * 65536ull;  // elems per time slice

// ---- feature detection for gfx1250 async-to-LDS path --------------------
#if defined(__HIP_DEVICE_COMPILE__) && defined(__gfx1250__)
  #if __has_builtin(__builtin_amdgcn_global_load_async_to_lds_b128) && \
      __has_builtin(__builtin_amdgcn_s_wait_asynccnt)
    #define IF_USE_ASYNC 1
  #else
    #define IF_USE_ASYNC 0
    #warning "gfx1250 async-to-LDS builtins unavailable; using direct global loads"
  #endif
#else
  #define IF_USE_ASYNC 0
#endif

#if IF_USE_ASYNC
  #if __has_builtin(__builtin_amdgcn_s_wait_dscnt)
    #define WAIT_DS0() __builtin_amdgcn_s_wait_dscnt(0)
  #else
    #define WAIT_DS0() asm volatile("s_wait_dscnt 0x0" ::: "memory")
  #endif
#endif

__device__ __forceinline__ v4f if_step(v4f m, v4f& v) {
  v4f o;
  o.x = (m.x >= 1.0f) ? 1.0f : 0.0f;
  o.y = (m.y >= 1.0f) ? 1.0f : 0.0f;
  o.z = (m.z >= 1.0f) ? 1.0f : 0.0f;
  o.w = (m.w >= 1.0f) ? 1.0f : 0.0f;
  v = m - o;  // soft reset by subtraction (v_th == 1.0)
  return o;
}

__global__ __launch_bounds__(BLOCK) void if_scan_kernel(
    const float* __restrict__ x, float* __restrict__ y) {
  const unsigned tid = threadIdx.x;
  const unsigned long long base =
      ((unsigned long long)blockIdx.x * BLOCK + tid) * VEC;
  const float* xp = x + base;
  float*       yp = y + base;

#if IF_USE_ASYNC
  // 8-deep async pipeline: prefetch slice t+7 while slice t is consumed from
  // LDS. Each lane loads and reads back only its own 16B slot, so per-wave
  // s_wait_asynccnt ordering suffices (no block barrier). The explicit
  // s_wait_dscnt 0 before each refill makes the slot WAR architecturally
  // safe (async LDS writes are unordered vs DS reads, ISA 08 §4.1); in
  // steady state DScnt is already 0, so it's a free SOPP.
  __shared__ __align__(16) float buf[DEPTH][BLOCK * VEC];

  // generic->AS3: low 32 bits of a generic LDS address are the LDS offset
  // (flat->LDS truncation, ISA 10.2); int->pointer cast is always accepted.
  #define ASYNC_LD(idx)                                                      \
    __builtin_amdgcn_global_load_async_to_lds_b128(                          \
        (v4i_gcc*)(xp + (unsigned long long)(idx) * BN),                     \
        (lds_v4i*)(unsigned int)(unsigned long long)                         \
            &buf[(idx) & (DEPTH - 1)][tid * VEC],                            \
        0, 0)

  #pragma unroll
  for (int p = 0; p < DEPTH - 1; ++p) ASYNC_LD(p);

  v4f v = {0.0f, 0.0f, 0.0f, 0.0f};
  #pragma unroll
  for (int t = 0; t < T_STEPS; ++t) {
    if (t + (DEPTH - 1) < T_STEPS) {
      WAIT_DS0();               // prior ds_load of this slot fully retired
      ASYNC_LD(t + (DEPTH - 1));
    }
    // wait until slice t has landed in LDS (async loads complete in order);
    // builtin needs a literal immediate -> constant-folded switch
    const int rem  = T_STEPS - 1 - t;
    const int need = rem < (DEPTH - 1) ? rem : (DEPTH - 1);
    switch (need) {
      case 0:  __builtin_amdgcn_s_wait_asynccnt(0); break;
      case 1:  __builtin_amdgcn_s_wait_asynccnt(1); break;
      case 2:  __builtin_amdgcn_s_wait_asynccnt(2); break;
      case 3:  __builtin_amdgcn_s_wait_asynccnt(3); break;
      case 4:  __builtin_amdgcn_s_wait_asynccnt(4); break;
      case 5:  __builtin_amdgcn_s_wait_asynccnt(5); break;
      case 6:  __builtin_amdgcn_s_wait_asynccnt(6); break;
      default: __builtin_amdgcn_s_wait_asynccnt(7); break;
    }
    v4f m = *(const v4f*)&buf[t & (DEPTH - 1)][tid * VEC];  // ds_load_b128
    m += v;
    v4f o = if_step(m, v);
    __builtin_nontemporal_store(
        o, (v4f*)(yp + (unsigned long long)t * BN));        // streaming store
  }
  #undef ASYNC_LD
#else
  // Fallback: direct coalesced b128 streaming loads, unrolled for MLP.
  v4f v = {0.0f, 0.0f, 0.0f, 0.0f};
  #pragma unroll 8
  for (int t = 0; t < T_STEPS; ++t) {
    v4f m = __builtin_nontemporal_load(
        (const v4f*)(xp + (unsigned long long)t * BN));
    m += v;
    v4f o = if_step(m, v);
    __builtin_nontemporal_store(
        o, (v4f*)(yp + (unsigned long long)t * BN));
  }
#endif
}

extern "C" void kernel_launch(void* const* d_in, const int* in_sizes, int n_in,
                              void* d_out, int out_size, void* d_ws,
                              size_t ws_size, hipStream_t stream) {
  const float* x = (const float*)d_in[0];
  float*       y = (float*)d_out;
  // B*N columns / (BLOCK threads * VEC cols per thread) = 2048 blocks
  const unsigned grid = (unsigned)(BN / (BLOCK * VEC));
  if_scan_kernel<<<grid, BLOCK, 0, stream>>>(x, y);
}